// YoloLayer_68075231641844
// MI455X (gfx1250) — compile-verified
//
#include <hip/hip_runtime.h>
#include <math.h>

// ---------------- problem constants (from reference setup) ----------------
#define NB   32
#define NA   3
#define NH   52
#define NW   52
#define NC   80
#define SPATIAL      (NH * NW)        // 2704
#define CELLS_PER_B  (NA * SPATIAL)   // 8112
#define MAXB 50
#define IGNORE_T 0.5f

// masked anchors: (10,13),(16,30),(33,23) / 32
__constant__ float c_aw[3] = {10.f / 32.f, 16.f / 32.f, 33.f / 32.f};
__constant__ float c_ah[3] = {13.f / 32.f, 30.f / 32.f, 23.f / 32.f};

__device__ __forceinline__ float sigmoidf(float x) {
    return 1.f / (1.f + expf(-x));
}
// torch-style clamped log used by bce_sum
__device__ __forceinline__ float clog(float p) {
    return fmaxf(logf(fmaxf(p, 1e-38f)), -100.f);
}
// IoU on (cx,cy,w,h) boxes, matching iou_cxcywh exactly
__device__ __forceinline__ float iou_cc(float px, float py, float pw, float ph,
                                        float gx, float gy, float gw, float gh) {
    float mx = fminf(px - pw * 0.5f, gx - gw * 0.5f);
    float Mx = fmaxf(px + pw * 0.5f, gx + gw * 0.5f);
    float my = fminf(py - ph * 0.5f, gy - gh * 0.5f);
    float My = fmaxf(py + ph * 0.5f, gy + gh * 0.5f);
    float cw = pw + gw - (Mx - mx);
    float ch = ph + gh - (My - my);
    if (cw <= 0.f || ch <= 0.f) return 0.f;
    float ca = cw * ch;
    return ca / (pw * ph + gw * gh - ca);
}

// ---------------------------------------------------------------------------
// Main fused kernel: one thread per (a,j,i) cell of one batch image.
// Builds the 50-entry GT table in LDS, then each cell does the 50-IoU loop,
// resolving obj / last-write-wins target fields + class-bit union inline.
// Only 5 of 85 channels are read per cell; the 80 class channels are read
// only at obj cells (the reference's cls loss is exactly 0 elsewhere).
// Block partials are written to d_ws (deterministic; no float atomics).
// ---------------------------------------------------------------------------
__global__ __launch_bounds__(256) void yolo_main(const float* __restrict__ outp,
                                                 const float* __restrict__ tgt,
                                                 float4* __restrict__ partials) {
    __shared__ float gt[MAXB * 16];
    __shared__ float red0[256], red1[256], red2[256];

    const int b   = blockIdx.y;
    const int tid = threadIdx.x;

    // ---- build per-batch GT table (threads 0..49) ----
    if (tid < MAXB) {
        const float* tb = tgt + b * (MAXB * 5) + tid * 5;
        float cls = tb[0], cx = tb[1], cy = tb[2], w = tb[3], h = tb[4];
        // cumulative validity: all t' <= tid must have cx != 0
        int v = 1;
        for (int u = 0; u <= tid; ++u)
            v &= (tgt[b * (MAXB * 5) + u * 5 + 1] != 0.0f);
        float gx = cx * NW, gy = cy * NH, gw = w * NW, gh = h * NH;
        // best anchor: zero-centered IoU => intersection = min(w)*min(h)
        int bn = 0; float best = -1.f;
        for (int a = 0; a < 3; ++a) {
            float cw = fminf(gw, c_aw[a]);
            float ch = fminf(gh, c_ah[a]);
            float ca = (cw <= 0.f || ch <= 0.f) ? 0.f : cw * ch;
            float iou = ca / (gw * gh + c_aw[a] * c_ah[a] - ca);
            if (iou > best) { best = iou; bn = a; }   // first-max like argmax
        }
        float fgi = floorf(gx), fgj = floorf(gy);
        int gi = min(max((int)fgi, 0), NW - 1);
        int gj = min(max((int)fgj, 0), NH - 1);
        float* g = gt + tid * 16;
        g[0] = gx; g[1] = gy; g[2] = gw; g[3] = gh;
        g[4] = v ? 1.f : 0.f;
        g[5] = (float)bn; g[6] = (float)gi; g[7] = (float)gj;
        g[8] = gx - fgi;  g[9] = gy - fgj;
        g[10] = logf(gw / c_aw[bn]);
        g[11] = logf(gh / c_ah[bn]);
        g[12] = sqrtf(2.f - w * h);
        g[13] = (float)((int)cls);
        g[14] = 0.f; g[15] = 0.f;
    }
    __syncthreads();

    const int cell = blockIdx.x * blockDim.x + tid;
    float lcoord = 0.f, lconf = 0.f, lcls = 0.f;

    if (cell < CELLS_PER_B) {
        const int a   = cell / SPATIAL;
        const int rem = cell - a * SPATIAL;
        const int j   = rem / NW;
        const int i   = rem - j * NW;

        const float* base = outp + ((size_t)b * (NA * 85) + a * 85) * SPATIAL + rem;
        float sx   = sigmoidf(base[0]);
        float sy   = sigmoidf(base[SPATIAL]);
        float rw   = base[2 * SPATIAL];
        float rh   = base[3 * SPATIAL];
        float conf = sigmoidf(base[4 * SPATIAL]);

        // reference's flattened anchor quirk for pred boxes
        int aidx = ((b * NA + a) * SPATIAL + rem) / (NB * SPATIAL);
        float px = sx + (float)i, py = sy + (float)j;
        float pw = expf(rw) * c_aw[aidx];
        float ph = expf(rh) * c_ah[aidx];

        float maxiou = 0.f;
        int   obj = 0;
        float sc = 0.f, t0 = 0.f, t1 = 0.f, t2 = 0.f, t3 = 0.f;
        unsigned cm0 = 0u, cm1 = 0u, cm2 = 0u;

        for (int t = 0; t < MAXB; ++t) {
            const float* g = gt + t * 16;
            if (g[4] == 0.f) continue;          // invalid (uniform per block)
            maxiou = fmaxf(maxiou,
                           iou_cc(px, py, pw, ph, g[0], g[1], g[2], g[3]));
            if ((int)g[5] == a && (int)g[6] == i && (int)g[7] == j) {
                obj = 1;                         // last-write-wins (ascending t)
                sc = g[12]; t0 = g[8]; t1 = g[9]; t2 = g[10]; t3 = g[11];
                int c = (int)g[13];
                if (c < 32)      cm0 |= 1u << c;
                else if (c < 64) cm1 |= 1u << (c - 32);
                else             cm2 |= 1u << (c - 64);
            }
        }

        float back = (maxiou > IGNORE_T) ? 0.f : 1.f;
        if (obj) back = 0.f;

        // conf loss:  bce(conf*back, 0) + bce(conf*objm, objm)
        {
            float p = conf * back;
            lconf = -clog(1.f - p);             // exactly 0 when back == 0
            if (obj) lconf += -clog(conf);
        }

        if (obj) {
            // coord loss: 0.5 * sum (coord*mask - tcoord*mask)^2
            float d0 = sx * sc - t0 * sc;
            float d1 = sy * sc - t1 * sc;
            float d2 = rw * sc - t2 * sc;
            float d3 = rh * sc - t3 * sc;
            lcoord = 0.5f * (d0 * d0 + d1 * d1 + d2 * d2 + d3 * d3);

            // class loss: only obj cells contribute (p = sigmoid*objm)
            const float* cb = base + 5 * SPATIAL;
            __builtin_prefetch(cb, 0, 0);                         // global_prefetch_b8
            __builtin_prefetch(cb + 40 * SPATIAL, 0, 0);
            for (int c = 0; c < NC; ++c) {
                float p = sigmoidf(cb[c * SPATIAL]);
                unsigned bit = (c < 32)  ? (cm0 >> c) & 1u
                             : (c < 64)  ? (cm1 >> (c - 32)) & 1u
                                         : (cm2 >> (c - 64)) & 1u;
                float tv = (float)bit;
                lcls += -(tv * clog(p) + (1.f - tv) * clog(1.f - p));
            }
        }
    }

    // ---- deterministic block reduction ----
    red0[tid] = lcoord; red1[tid] = lconf; red2[tid] = lcls;
    __syncthreads();
    for (int s = 128; s > 0; s >>= 1) {
        if (tid < s) {
            red0[tid] += red0[tid + s];
            red1[tid] += red1[tid + s];
            red2[tid] += red2[tid + s];
        }
        __syncthreads();
    }
    if (tid == 0)
        partials[blockIdx.y * gridDim.x + blockIdx.x] =
            make_float4(red0[0], red1[0], red2[0], 0.f);
}

// ---------------------------------------------------------------------------
// Final reduction: one wave32. Each lane serially accumulates a fixed subset
// of block partials into two fp32 accumulators, then one
// V_WMMA_F32_16X16X4_F32 per loss component sums all 64 accumulators on the
// matrix unit (B = all-ones -> exact fp32 sum, fixed HW order, deterministic).
// ---------------------------------------------------------------------------
typedef float v2f __attribute__((ext_vector_type(2)));
typedef float v8f __attribute__((ext_vector_type(8)));

__global__ __launch_bounds__(32) void yolo_reduce(const float4* __restrict__ partials,
                                                  float* __restrict__ outv, int n) {
    const int lane = threadIdx.x;
    float ax[3] = {0.f, 0.f, 0.f};
    float ay[3] = {0.f, 0.f, 0.f};
    for (int p = lane; p < n; p += 32) {
        float4 v = partials[p];
        if (((p >> 5) & 1) == 0) { ax[0] += v.x; ax[1] += v.y; ax[2] += v.z; }
        else                     { ay[0] += v.x; ay[1] += v.y; ay[2] += v.z; }
    }
#if defined(__has_builtin) && __has_builtin(__builtin_amdgcn_wmma_f32_16x16x4_f32)
    v2f bones; bones.x = 1.f; bones.y = 1.f;        // B = ones => D row m = sum_k A[m][k]
    for (int c = 0; c < 3; ++c) {
        v2f A; A.x = ax[c]; A.y = ay[c];            // A 16x4 = 64 lane accumulators
        v8f C = {};
        C = __builtin_amdgcn_wmma_f32_16x16x4_f32(
                false, A, false, bones, (short)0, C, false, false);
        // lanes 0-15 hold rowsums M=0..7 in C[0..7]; lanes 16-31 hold M=8..15
        float h = C[0] + C[1] + C[2] + C[3] + C[4] + C[5] + C[6] + C[7];
        float tot = __shfl(h, 0, 32) + __shfl(h, 16, 32);
        if (lane == 0) outv[c] = tot / (float)NB;
    }
#else
    for (int c = 0; c < 3; ++c) {
        float s = ax[c] + ay[c];
        for (int off = 16; off > 0; off >>= 1) s += __shfl_down(s, off, 32);
        if (lane == 0) outv[c] = s / (float)NB;
    }
#endif
}

// ---------------------------------------------------------------------------
extern "C" void kernel_launch(void* const* d_in, const int* in_sizes, int n_in,
                              void* d_out, int out_size, void* d_ws, size_t ws_size,
                              hipStream_t stream) {
    const float* outp = (const float*)d_in[0];   // (32,255,52,52) fp32
    const float* tgt  = (const float*)d_in[1];   // (32,250)       fp32
    float4* partials  = (float4*)d_ws;           // 1024 * 16 B = 16 KB scratch

    dim3 grid((CELLS_PER_B + 255) / 256, NB);    // 32 x 32 blocks
    yolo_main<<<grid, 256, 0, stream>>>(outp, tgt, partials);

    int nblocks = grid.x * grid.y;               // 1024
    yolo_reduce<<<1, 32, 0, stream>>>(partials, (float*)d_out, nblocks);
}